// TripletLoss_43662637531717
// MI455X (gfx1250) — compile-verified
//
#include <hip/hip_runtime.h>
#include <math.h>

#define NN 512
#define DD 64
#define CC 32

typedef __attribute__((ext_vector_type(2))) float v2f;
typedef __attribute__((ext_vector_type(8))) float v8f;

// ---------------------------------------------------------------------------
// Prep: row squared-norms of u, and integer labels from one-hot y.
// ---------------------------------------------------------------------------
__global__ void __launch_bounds__(256) tl_prep(const float* __restrict__ u,
                                               const float* __restrict__ y,
                                               float* __restrict__ norms,
                                               int* __restrict__ labels) {
  int i = blockIdx.x * blockDim.x + threadIdx.x;
  if (i >= NN) return;
  float s = 0.f;
  #pragma unroll 8
  for (int k = 0; k < DD; ++k) {
    float v = u[i * DD + k];
    s += v * v;
  }
  norms[i] = s;
  int lab = 0;
  #pragma unroll
  for (int c = 0; c < CC; ++c) lab = (y[i * CC + c] > 0.5f) ? c : lab;
  labels[i] = lab;
}

// ---------------------------------------------------------------------------
// Distance matrix via fp32 WMMA Gram tiles:
//   sq[i][j] = |u_i|^2 + |u_j|^2 - 2 * (U U^T)[i][j];  d = sq>0 ? sqrt : 0
// One wave per 16x16 tile; V_WMMA_F32_16X16X4_F32, K-loop over D=64.
// A layout (fp32 16x4): lane m=lane%16 holds row i0+m, K pair at 2*(lane/16).
// B layout mirrors with N; C/D: VGPR v -> (M = v + 8*(lane/16), N = lane%16).
// ---------------------------------------------------------------------------
__global__ void __launch_bounds__(256) tl_dist(const float* __restrict__ u,
                                               const float* __restrict__ norms,
                                               float* __restrict__ dmat) {
  const int lane = threadIdx.x & 31;
  const int wave = threadIdx.x >> 5;
  const int w  = blockIdx.x * 8 + wave;   // global wave = tile id
  const int ti = w >> 5;                  // 32 tiles per row
  const int tj = w & 31;
  const int i0 = ti * 16, j0 = tj * 16;
  const int half = lane >> 4;
  const int mn   = lane & 15;

  v8f c = {};
  const float* arow = u + (i0 + mn) * DD + 2 * half;
  const float* brow = u + (j0 + mn) * DD + 2 * half;
  #pragma unroll
  for (int kk = 0; kk < DD; kk += 4) {
    v2f a = *(const v2f*)(arow + kk);
    v2f b = *(const v2f*)(brow + kk);
    // 8 args: (neg_a, A, neg_b, B, c_mod, C, reuse_a, reuse_b)
    c = __builtin_amdgcn_wmma_f32_16x16x4_f32(false, a, false, b,
                                              (short)0, c, false, false);
  }

  const float nj = norms[j0 + mn];
  #pragma unroll
  for (int v = 0; v < 8; ++v) {
    int row = i0 + v + 8 * half;
    int col = j0 + mn;
    float sq = norms[row] + nj - 2.0f * c[v];
    float d  = (row != col && sq > 0.f) ? sqrtf(sq) : 0.f;
    dmat[row * NN + col] = d;
  }
}

// ---------------------------------------------------------------------------
// Semi-hard triplet mining + hinge sum. One block per anchor row.
// row_valid is implied: diagonal gives any(s); has_neg gives any(~s).
// ---------------------------------------------------------------------------
__global__ void __launch_bounds__(256) tl_triplet(const float* __restrict__ dmat,
                                                  const int* __restrict__ labels,
                                                  const int* __restrict__ alphaRaw,
                                                  float* __restrict__ out) {
  __shared__ float sd[NN];
  __shared__ int   sl[NN];
  __shared__ float red[256];

  const int i   = blockIdx.x;
  const int tid = threadIdx.x;

  // ALPHA dtype hedge: small integer -> int payload; otherwise float bits.
  int av = *alphaRaw;
  float alpha = (av > -16777216 && av < 16777216) ? (float)av
                                                  : __int_as_float(av);

  for (int t = tid; t < NN; t += 256) {
    sd[t] = dmat[i * NN + t];
    sl[t] = labels[t];
  }
  __syncthreads();

  const int li = sl[i];
  float acc = 0.f;
  for (int j = tid; j < NN; j += 256) {
    if (sl[j] != li) continue;          // positive requires same class
    float dij = sd[j];
    if (dij == 0.f) continue;           // excludes diagonal / duplicates
    float hi   = dij + alpha;
    float nmin = 1e30f;
    bool  hasneg = false;
    for (int k = 0; k < NN; ++k) {
      float dik = sd[k];
      bool cand = (sl[k] != li) && (dij < dik) && (dik < hi);
      if (cand) { nmin = fminf(nmin, dik); hasneg = true; }
    }
    if (hasneg) acc += fmaxf(dij - nmin + alpha, 0.f);
  }

  red[tid] = acc;
  __syncthreads();
  #pragma unroll
  for (int s = 128; s > 0; s >>= 1) {
    if (tid < s) red[tid] += red[tid + s];
    __syncthreads();
  }
  if (tid == 0) atomicAdd(out, red[0]);
}

// ---------------------------------------------------------------------------
extern "C" void kernel_launch(void* const* d_in, const int* in_sizes, int n_in,
                              void* d_out, int out_size, void* d_ws, size_t ws_size,
                              hipStream_t stream) {
  (void)in_sizes; (void)n_in; (void)out_size; (void)ws_size;
  const float* u     = (const float*)d_in[0];
  const float* y     = (const float*)d_in[1];
  const int*   alpha = (const int*)d_in[2];
  float*       out   = (float*)d_out;

  float* dmat   = (float*)d_ws;          // 512*512 f32 = 1 MB
  float* norms  = dmat + NN * NN;        // 512 f32
  int*   labels = (int*)(norms + NN);    // 512 i32

  tl_prep<<<2, 256, 0, stream>>>(u, y, norms, labels);
  tl_dist<<<(NN / 16) * (NN / 16) / 8, 256, 0, stream>>>(u, norms, dmat);
  hipMemsetAsync(d_out, 0, sizeof(float), stream);
  tl_triplet<<<NN, 256, 0, stream>>>(dmat, labels, alpha, out);
}